// RedistributionNetwork_49495203119752
// MI455X (gfx1250) — compile-verified
//
#include <hip/hip_runtime.h>
#include <stdint.h>

// ---------------------------------------------------------------------------
// RedistributionNetwork: x = x @ W[l].T for l in 0..127, x:[256,1024] f32.
//
// MI455X plan:
//  * persistent grid of 128 WGs (8 N-groups x 16 B-groups), 8 waves each
//  * per wave per layer: one 16x16 output tile via V_WMMA_F32_16X16X4_F32,
//    K=1024 in 256 steps (exact f32 math)
//  * h tile [16 x 1024] staged to LDS once/layer via TDM tensor_load_to_lds
//    with descriptor padding (2 DW per 256 DW -> stride 1032 w: 8B-aligned
//    ds_load_b64, 2-way instead of 16-way bank conflicts)
//  * W streamed from global (rows contiguous, L2-shared across B-groups:
//    537 MB total HBM traffic, ~23 us floor at 23.3 TB/s)
//  * software grid barrier between layers (atomic counter per layer in d_ws)
// ---------------------------------------------------------------------------

#define NXL   128          // layers
#define NYV   1024         // state width
#define BATCH 256          // batch rows
#define NBLK  128          // 8 n-groups * 16 b-groups (persistent)
#define TPB   256          // 8 wave32
#define LDS_ROW 1032       // 1024 + 2 pad DWORDs per 256 DWORDs (TDM pad)
#define LDS_WORDS (16 * LDS_ROW + 16)

typedef float    v2f __attribute__((ext_vector_type(2)));
typedef float    v4f __attribute__((ext_vector_type(4)));
typedef float    v8f __attribute__((ext_vector_type(8)));
typedef uint32_t v4u __attribute__((ext_vector_type(4)));
typedef int      v4i __attribute__((ext_vector_type(4)));
typedef int      v8i __attribute__((ext_vector_type(8)));

__global__ void redist_init_cnt(uint32_t* c) {
  if (threadIdx.x < NXL) c[threadIdx.x] = 0u;   // re-zeroed every launch
}

__global__ __launch_bounds__(TPB)
void redist_chain_kernel(const float* __restrict__ x,
                         const float* __restrict__ W,
                         float* __restrict__ out,
                         float* __restrict__ hbuf,
                         uint32_t* __restrict__ cnt)
{
  __shared__ float h_lds[LDS_WORDS];

  const int tid  = threadIdx.x;
  const int lane = tid & 31;
  const int wave = tid >> 5;
  const int r    = lane & 15;        // tile row-within-16 (A: M, B/D: N)
  const int hsel = lane >> 4;        // half-wave select
  const int bg   = blockIdx.x & 15;  // batch group (16 rows)
  const int ng   = blockIdx.x >> 4;  // n group (128 cols)
  const int bbase  = bg * 16;
  const int n_base = ng * 128 + wave * 16;
  const int nrow   = n_base + r;     // W row this lane streams (A operand)

  const uint32_t lds_base = (uint32_t)(uintptr_t)(&h_lds[0]); // LDS byte offset

  for (int l = 0; l < NXL; ++l) {
    const float* src = (l == 0) ? x : (((l & 1) != 0) ? hbuf : out);
    float*       dst = (((l + 1) & 1) != 0) ? hbuf : out;   // l==127 -> out

    // ---- TDM: DMA h tile rows [bbase, bbase+16) x 1024 f32 into LDS,
    //      inserting 2 pad DWORDs after every 256 DWORDs (row stride 1032 w).
    if (wave == 0) {
      uint64_t ga = (uint64_t)(uintptr_t)(src + (size_t)bbase * NYV);
      v4u g0;
      g0[0] = 1u;                                        // count=1 (valid D#)
      g0[1] = lds_base;                                  // lds_addr (bytes)
      g0[2] = (uint32_t)ga;                              // global_addr[31:0]
      g0[3] = (uint32_t)((ga >> 32) & 0x1FFFFFFull)      // global_addr[56:32]
            | (2u << 30);                                // type=2 ("image")
      v8i g1;
      g1[0] = (2 << 16)      // data_size = 4B
            | (1 << 20)      // pad_enable
            | (7 << 22)      // pad_interval: 256 DWORDs
            | (1 << 25);     // pad_amount: 2 DWORDs
      g1[1] = (int)(1024u << 16);  // tensor_dim0 lo16 = 1024 (atomic_barrier=0)
      g1[2] = (int)(256u << 16);   // tensor_dim0 hi16=0 | tensor_dim1 lo16=256
      g1[3] = (int)(1024u << 16);  // tensor_dim1 hi16=0 | tile_dim0 = 1024
      g1[4] = 16;                  // tile_dim1 = 16, tile_dim2 = 0
      g1[5] = 1024;                // tensor_dim0_stride lo32 = 1024 elements
      g1[6] = 0;
      g1[7] = 0;
      v4i gz4 = {0, 0, 0, 0};      // 2D tile: groups 2/3 unused
      v8i gz8 = {0, 0, 0, 0, 0, 0, 0, 0};
      __builtin_amdgcn_tensor_load_to_lds(g0, g1, gz4, gz4, gz8, 0);
      __builtin_amdgcn_s_wait_tensorcnt(0);
    }
    __syncthreads();

    // ---- D[n,b] = sum_k W[l][n][k] * h[b][k]  (out.T = W @ h.T)
    // A frag (16x4 f32): lane(r,hsel), vgpr j -> W[nrow][k0 + 2*hsel + j]
    // B frag (4x16 f32): lane(r,hsel), vgpr j -> h[bbase+r][k0 + 2*hsel + j]
    const float* wp = W + ((size_t)l << 20) + (size_t)nrow * NYV + 2 * hsel;
    const float* bp = &h_lds[r * LDS_ROW + 2 * hsel];

    v8f acc = {0.f, 0.f, 0.f, 0.f, 0.f, 0.f, 0.f, 0.f};
    #pragma unroll 8
    for (int k0 = 0; k0 < NYV; k0 += 4) {
      v2f a = *(const v2f*)(wp + k0);                       // global_load_b64
      v2f b = *(const v2f*)(bp + k0 + 2 * (k0 >> 8));       // ds_load_b64
      acc = __builtin_amdgcn_wmma_f32_16x16x4_f32(
                false, a, false, b, (short)0, acc, false, false);
    }

    // ---- D store: vgpr q, lane(r,hsel) = out[b = bbase+r][n = n_base+q+8*hsel]
    float* dp = dst + (size_t)(bbase + r) * NYV + n_base + 8 * hsel;
    v4f lo = {acc[0], acc[1], acc[2], acc[3]};
    v4f hi = {acc[4], acc[5], acc[6], acc[7]};
    *(v4f*)(dp)     = lo;                                   // global_store_b128
    *(v4f*)(dp + 4) = hi;

    // ---- inter-layer grid barrier (all 128 WGs co-resident / persistent)
    if (l != NXL - 1) {
      __syncthreads();
      if (tid == 0) {
        __threadfence();
        __hip_atomic_fetch_add(&cnt[l], 1u, __ATOMIC_ACQ_REL,
                               __HIP_MEMORY_SCOPE_AGENT);
        while (__hip_atomic_load(&cnt[l], __ATOMIC_ACQUIRE,
                                 __HIP_MEMORY_SCOPE_AGENT) < (uint32_t)NBLK) {
          __builtin_amdgcn_s_sleep(1);
        }
      }
      __syncthreads();
    }
  }
}

extern "C" void kernel_launch(void* const* d_in, const int* in_sizes, int n_in,
                              void* d_out, int out_size, void* d_ws, size_t ws_size,
                              hipStream_t stream) {
  const float* x = (const float*)d_in[0];     // [256, 1024] f32
  const float* W = (const float*)d_in[1];     // [128, 1024, 1024] f32
  float* out = (float*)d_out;                 // [256, 1024] f32
  uint32_t* cnt = (uint32_t*)d_ws;            // 128 barrier counters
  float* hbuf = (float*)((char*)d_ws + 1024); // 1 MB ping-pong h buffer

  redist_init_cnt<<<1, 128, 0, stream>>>(cnt);
  redist_chain_kernel<<<NBLK, TPB, 0, stream>>>(x, W, out, hbuf, cnt);
}